// AFMLayer_68186900791340
// MI455X (gfx1250) — compile-verified
//
#include <hip/hip_runtime.h>

// AFM layer, reduced to its live computation.
// Reference math: softmax over a size-1 axis == 1.0, so the attention MLP
// (Wa, ba, Wh, bh) is dead code; pooled = sum over all 325 pairwise products
// = 0.5 * ((sum_f e_f)^2 - sum_f e_f^2)  (FM identity).
// out = sigmoid(pooled @ Wo + bo), fp32 throughout.

typedef __attribute__((ext_vector_type(2))) float v2f;
typedef __attribute__((ext_vector_type(8))) float v8f;

#define N_FIELDS 26
#define VOCAB    10000
#define KDIM     16
#define BATCH    2048
#define ROWS_PER_BLOCK 16   // one 16-row WMMA tile per block

__global__ __launch_bounds__(256) void afm_fused_kernel(
    const int*   __restrict__ sparse,   // [BATCH, 26]
    const float* __restrict__ emb,      // [26, 10000, 16]
    const float* __restrict__ Wo,       // [16]
    const float* __restrict__ bo,       // [1]
    float*       __restrict__ out)      // [BATCH, 1]
{
    __shared__ float pooled[ROWS_PER_BLOCK * KDIM];

    const int tid  = threadIdx.x;
    const int wave = tid >> 5;          // 8 waves of 32 (wave32)
    const int lane = tid & 31;
    const int half = lane >> 4;         // half-wave id
    const int k    = lane & 15;         // embedding component
    const int rowInTile = (wave << 1) | half;
    const int row  = blockIdx.x * ROWS_PER_BLOCK + rowInTile;

    // ---- Phase 1: gather + FM reduction. One row per 16 lanes; lane k reads
    // element k of each 64B embedding row -> coalesced 64B transactions.
    float S = 0.0f, Q = 0.0f;
    const int* srow = sparse + row * N_FIELDS;
    #pragma unroll
    for (int f = 0; f < N_FIELDS; ++f) {
        const int idx = srow[f];  // uniform across the 16 lanes; L0 absorbs
        const float v = emb[((size_t)f * VOCAB + (size_t)idx) * KDIM + k];
        S += v;
        Q = fmaf(v, v, Q);
    }
    pooled[rowInTile * KDIM + k] = 0.5f * (S * S - Q);
    __syncthreads();

    // ---- Phase 2: pooled[16,16] @ Wo[16,1] via V_WMMA_F32_16X16X4_F32,
    // accumulating 4 steps of K=4. Wave 0 only (branch is wave-uniform, so
    // EXEC is all 1s inside, as WMMA requires).
    if (wave == 0) {
        const int m = lane & 15;   // A-matrix row M = lane % 16
        const int h = lane >> 4;   // K-pair select: VGPR slot covers K=2h,2h+1

        v8f c = {};
        #pragma unroll
        for (int s = 0; s < 4; ++s) {
            // A layout (ISA 7.12.2, 32-bit A 16x4): lanes 0-15 hold K={0,1},
            // lanes 16-31 hold K={2,3}; logical k = 4s + 2h + {0,1}.
            v2f a, b;
            a.x = pooled[m * KDIM + 4 * s + 2 * h];
            a.y = pooled[m * KDIM + 4 * s + 2 * h + 1];
            // B 4x16: same K split; broadcast Wo[k] across all 16 N columns,
            // so every column of D equals pooled_row . Wo.
            const float2 w = ((const float2*)Wo)[2 * s + h];
            b.x = w.x;
            b.y = w.y;
            c = __builtin_amdgcn_wmma_f32_16x16x4_f32(
                    /*neg_a=*/false, a, /*neg_b=*/false, b,
                    /*c_mod=*/(short)0, c, /*reuse_a=*/false, /*reuse_b=*/false);
        }

        // D layout: VGPR j -> M = j (lanes 0-15) or M = j+8 (lanes 16-31),
        // identical across N. Lanes 0 and 16 each store 8 rows.
        const float bias = bo[0];
        if (m == 0) {
            #pragma unroll
            for (int j = 0; j < 8; ++j) {
                const float x = c[j] + bias;
                out[blockIdx.x * ROWS_PER_BLOCK + j + 8 * h] =
                    1.0f / (1.0f + __expf(-x));
            }
        }
    }
}

extern "C" void kernel_launch(void* const* d_in, const int* in_sizes, int n_in,
                              void* d_out, int out_size, void* d_ws, size_t ws_size,
                              hipStream_t stream) {
    (void)in_sizes; (void)n_in; (void)out_size; (void)d_ws; (void)ws_size;
    // setup_inputs order:
    // 0: dense   (unused)          1: sparse [2048,26] int32
    // 2: emb_tables [26,10000,16]  3: Wa (dead)  4: ba (dead)
    // 5: Wh (dead)                 6: bh (dead)
    // 7: Wo [16]                   8: bo [1]
    const int*   sparse = (const int*)  d_in[1];
    const float* emb    = (const float*)d_in[2];
    const float* Wo     = (const float*)d_in[7];
    const float* bo     = (const float*)d_in[8];
    float*       out    = (float*)d_out;

    dim3 grid(BATCH / ROWS_PER_BLOCK);   // 128 blocks
    dim3 block(256);                     // 8 wave32 waves
    afm_fused_kernel<<<grid, block, 0, stream>>>(sparse, emb, Wo, bo, out);
}